// RSRModel_6287832122044
// MI455X (gfx1250) — compile-verified
//
#include <hip/hip_runtime.h>

typedef __attribute__((ext_vector_type(2))) float v2f;
typedef __attribute__((ext_vector_type(8))) float v8f;

#define NROWS 2048
#define TSTEPS 60
#define DFEAT 6
#define HID 64

__device__ __forceinline__ v8f wmma4(v2f a, v2f b, v8f c) {
  // D = A(16x4 f32) * B(4x16 f32) + C(16x16 f32), exact fp32 matrix core op
  return __builtin_amdgcn_wmma_f32_16x16x4_f32(false, a, false, b, (short)0, c, false, false);
}

__device__ __forceinline__ float sigm(float x) { return 1.0f / (1.0f + __expf(-x)); }

// 16x16 gate tile: C = A(h tile rows, K=64) * B(W^T col slice) + bias
// Alds: [16][64] row-major h buffer; Bcol: &W[col*64] (W row-major (192,64))
__device__ __forceinline__ v8f gate_gemm(const float* Alds, const float* Bcol,
                                         float bias, int li, int hi) {
  v8f c;
  #pragma unroll
  for (int v = 0; v < 8; ++v) c[v] = bias;
  #pragma unroll
  for (int kt = 0; kt < 16; ++kt) {
    const int k0 = kt*4 + 2*hi;
    v2f a = *(const v2f*)&Alds[li*HID + k0];
    v2f b = *(const v2f*)&Bcol[k0];
    c = wmma4(a, b, c);
  }
  return c;
}

// layer-0 input gate: x_t(16x6) @ Wih0^T col slice + bias, plain FMAs
__device__ __forceinline__ v8f gi0_valu(const float* xl, const float* sWih0,
                                        float bias, int col, int hi) {
  v8f a;
  #pragma unroll
  for (int v = 0; v < 8; ++v) a[v] = bias;
  #pragma unroll
  for (int d = 0; d < DFEAT; ++d) {
    const float w = sWih0[col*DFEAT + d];
    #pragma unroll
    for (int v = 0; v < 8; ++v)
      a[v] = fmaf(xl[(v + 8*hi)*DFEAT + d], w, a[v]);
  }
  return a;
}

// ---------------------------------------------------------------------------
// GRU: 2 stacked layers fused over time. block = 128 threads = 4 waves, each
// wave owns 16 samples. Weights live in LDS (one cooperative copy), h state is
// double-buffered in LDS; all recurrence GEMMs are fp32 WMMA 16x16x4.
// ---------------------------------------------------------------------------
__global__ __launch_bounds__(128) void gru_kernel(
    const float* __restrict__ x,
    const float* __restrict__ Wih0, const float* __restrict__ Whh0,
    const float* __restrict__ bih0, const float* __restrict__ bhh0,
    const float* __restrict__ Wih1, const float* __restrict__ Whh1,
    const float* __restrict__ bih1, const float* __restrict__ bhh1,
    float* __restrict__ xh)
{
  extern __shared__ float smem[];
  float* sWhh0 = smem;                 // 192*64 = 12288
  float* sWih1 = sWhh0 + 12288;        // 12288
  float* sWhh1 = sWih1 + 12288;        // 12288
  float* sWih0 = sWhh1 + 12288;        // 192*6 = 1152
  float* sBias = sWih0 + 1152;         // 4*192 = 768
  float* wbase = sBias + 768;          // per-wave: 4*1024 h + 96 x = 4192

  const int tid  = threadIdx.x;
  const int wave = tid >> 5;
  const int lane = tid & 31;
  const int li   = lane & 15;   // A row / C column within 16-wide tile
  const int hi   = lane >> 4;   // selects K pair (A/B) and row half (C/D)
  const int sbase = blockIdx.x * 64 + wave * 16;

  // cooperative weight staging (once per block)
  for (int idx = tid; idx < 12288; idx += 128) {
    sWhh0[idx] = Whh0[idx];
    sWih1[idx] = Wih1[idx];
    sWhh1[idx] = Whh1[idx];
  }
  for (int idx = tid; idx < 1152; idx += 128) sWih0[idx] = Wih0[idx];
  for (int idx = tid; idx < 192; idx += 128) {
    sBias[idx]       = bih0[idx];
    sBias[192 + idx] = bhh0[idx];
    sBias[384 + idx] = bih1[idx];
    sBias[576 + idx] = bhh1[idx];
  }

  float* wl  = wbase + wave * 4192;
  float* h0c = wl;            // layer-0 h, current
  float* h0n = wl + 1024;     // layer-0 h, next
  float* h1c = wl + 2048;     // layer-1 h, current
  float* h1n = wl + 3072;     // layer-1 h, next
  float* xl  = wl + 4096;     // x_t stage [16][6]

  for (int idx = lane; idx < 16*HID; idx += 32) { h0c[idx] = 0.f; h1c[idx] = 0.f; }
  __syncthreads();

  const float* sbih0 = sBias;
  const float* sbhh0 = sBias + 192;
  const float* sbih1 = sBias + 384;
  const float* sbhh1 = sBias + 576;

  for (int t = 0; t < TSTEPS; ++t) {
    // stage x_t for this wave's 16 samples: xs[n,t,d] = x[n, d*T + t]
    for (int it = lane; it < 16*DFEAT; it += 32) {
      int s = it / DFEAT, d = it - s*DFEAT;
      xl[it] = x[(size_t)(sbase + s) * (DFEAT*TSTEPS) + d*TSTEPS + t];
    }
    __syncthreads();

    // -------- layer 0: reads h0c, writes h0n --------
    #pragma unroll
    for (int ct = 0; ct < 4; ++ct) {
      const int colR = ct*16 + li;        // r gate column
      const int colZ = 64 + colR;         // z gate column
      const int colN = 128 + colR;        // n gate column

      v8f gi = gi0_valu(xl, sWih0, sbih0[colR], colR, hi);
      v8f gh = gate_gemm(h0c, sWhh0 + colR*HID, sbhh0[colR], li, hi);
      v8f rv;
      #pragma unroll
      for (int v = 0; v < 8; ++v) rv[v] = sigm(gi[v] + gh[v]);

      gi = gi0_valu(xl, sWih0, sbih0[colZ], colZ, hi);
      gh = gate_gemm(h0c, sWhh0 + colZ*HID, sbhh0[colZ], li, hi);
      v8f zv;
      #pragma unroll
      for (int v = 0; v < 8; ++v) zv[v] = sigm(gi[v] + gh[v]);

      gi = gi0_valu(xl, sWih0, sbih0[colN], colN, hi);
      gh = gate_gemm(h0c, sWhh0 + colN*HID, sbhh0[colN], li, hi);
      #pragma unroll
      for (int v = 0; v < 8; ++v) {
        const float n    = tanhf(gi[v] + rv[v] * gh[v]);
        const float hold = h0c[(v + 8*hi)*HID + ct*16 + li];
        h0n[(v + 8*hi)*HID + ct*16 + li] = (1.f - zv[v])*n + zv[v]*hold;
      }
    }
    __syncthreads();

    // -------- layer 1: reads h0n (input) and h1c, writes h1n --------
    #pragma unroll
    for (int ct = 0; ct < 4; ++ct) {
      const int colR = ct*16 + li;
      const int colZ = 64 + colR;
      const int colN = 128 + colR;

      v8f gi = gate_gemm(h0n, sWih1 + colR*HID, sbih1[colR], li, hi);
      v8f gh = gate_gemm(h1c, sWhh1 + colR*HID, sbhh1[colR], li, hi);
      v8f rv;
      #pragma unroll
      for (int v = 0; v < 8; ++v) rv[v] = sigm(gi[v] + gh[v]);

      gi = gate_gemm(h0n, sWih1 + colZ*HID, sbih1[colZ], li, hi);
      gh = gate_gemm(h1c, sWhh1 + colZ*HID, sbhh1[colZ], li, hi);
      v8f zv;
      #pragma unroll
      for (int v = 0; v < 8; ++v) zv[v] = sigm(gi[v] + gh[v]);

      gi = gate_gemm(h0n, sWih1 + colN*HID, sbih1[colN], li, hi);
      gh = gate_gemm(h1c, sWhh1 + colN*HID, sbhh1[colN], li, hi);
      #pragma unroll
      for (int v = 0; v < 8; ++v) {
        const float n    = tanhf(gi[v] + rv[v] * gh[v]);
        const float hold = h1c[(v + 8*hi)*HID + ct*16 + li];
        h1n[(v + 8*hi)*HID + ct*16 + li] = (1.f - zv[v])*n + zv[v]*hold;
      }
    }
    __syncthreads();

    // swap double buffers
    float* tp;
    tp = h0c; h0c = h0n; h0n = tp;
    tp = h1c; h1c = h1n; h1n = tp;
  }

  // x_hidden = final layer-1 hidden state (coalesced copy out)
  for (int idx = lane; idx < 16*HID; idx += 32)
    xh[(size_t)sbase*HID + idx] = h1c[idx];
}

// ---------------------------------------------------------------------------
// ai = xh @ W1, aj = xh @ W2, plus transposed copy xhT[64][2048] so the
// attention GEMM's B fragments are contiguous 8-byte loads.
// ---------------------------------------------------------------------------
__global__ __launch_bounds__(256) void attn_scalar_kernel(
    const float* __restrict__ xh, const float* __restrict__ W,
    float* __restrict__ ai, float* __restrict__ aj, float* __restrict__ xhT)
{
  const int i = blockIdx.x * blockDim.x + threadIdx.x;
  if (i >= NROWS) return;
  float s1 = 0.f, s2 = 0.f;
  for (int c = 0; c < HID; ++c) {
    const float h = xh[(size_t)i*HID + c];
    s1 = fmaf(h, W[c], s1);
    s2 = fmaf(h, W[HID + c], s2);
    xhT[(size_t)c*NROWS + i] = h;
  }
  ai[i] = s1;
  aj[i] = s2;
}

// ---------------------------------------------------------------------------
// Fused relation pass + masked leaky + softmax + (P @ x_hidden) + FC.
// 128 blocks x 256 threads (8 waves); 16 rows/block; dynamic LDS row buffer.
// Single streaming float4 read of the 335 MB relation tensor (HBM roofline).
// ---------------------------------------------------------------------------
__global__ __launch_bounds__(256) void attn_kernel(
    const float* __restrict__ rm,
    const float* __restrict__ xh,
    const float* __restrict__ xhT,
    const float* __restrict__ ai,
    const float* __restrict__ aj,
    const float* __restrict__ W,
    const float* __restrict__ b,
    const float* __restrict__ fcw,
    const float* __restrict__ fcb,
    float* __restrict__ out)
{
  extern __shared__ float smem[];
  float* mw      = smem;                    // [16][2048] masked weights / probs
  float* red     = mw + 16*2048;            // [256] reduction scratch
  float* rowstat = red + 256;               // [16] row max, then row sum
  float* hidbuf  = rowstat + 16;            // [16][64]
  float* partial = hidbuf + 16*HID;         // [4][256] K-half partials

  const int tid = threadIdx.x;
  const int i0  = blockIdx.x * 16;

  {
    const int r  = tid >> 4;                // row 0..15
    const int jg = tid & 15;                // 16 threads sweep j per row
    const int i  = i0 + r;
    float w3[20];
    #pragma unroll
    for (int q = 0; q < 20; ++q) w3[q] = W[2*HID + q];
    const float av   = ai[i];
    const float bias = b[0];

    // phase 1: ar-dot + mask-sum in one float4 pass over rm; row max
    float lmax = -3.4e38f;
    for (int jj = 0; jj < 128; ++jj) {
      const int j = jj*16 + jg;
      const float4* p4 = (const float4*)(rm + ((size_t)i * NROWS + j) * 20);
      __builtin_prefetch((const float*)p4 + 16*20, 0, 0);  // next line of 16 j's
      float s = 0.f, d = 0.f;
      #pragma unroll
      for (int q4 = 0; q4 < 5; ++q4) {
        const float4 vv = p4[q4];
        s += vv.x + vv.y + vv.z + vv.w;
        d = fmaf(vv.x, w3[4*q4+0], d);
        d = fmaf(vv.y, w3[4*q4+1], d);
        d = fmaf(vv.z, w3[4*q4+2], d);
        d = fmaf(vv.w, w3[4*q4+3], d);
      }
      float w = av + aj[j] + d + bias;
      w = (w >= 0.f) ? w : 0.01f * w;        // leaky relu
      const float m = (s != 0.f) ? w : 0.f;  // mask * weight
      mw[r*2048 + j] = m;
      lmax = fmaxf(lmax, m);
    }
    red[tid] = lmax;
    __syncthreads();
    if (jg == 0) {
      float m = red[r*16];
      #pragma unroll
      for (int k = 1; k < 16; ++k) m = fmaxf(m, red[r*16 + k]);
      rowstat[r] = m;
    }
    __syncthreads();

    // phase 2: exp + row sum
    const float rmax = rowstat[r];
    float lsum = 0.f;
    for (int jj = 0; jj < 128; ++jj) {
      const int j = jj*16 + jg;
      const float e = __expf(mw[r*2048 + j] - rmax);
      mw[r*2048 + j] = e;
      lsum += e;
    }
    red[tid] = lsum;
    __syncthreads();
    if (jg == 0) {
      float s = 0.f;
      #pragma unroll
      for (int k = 0; k < 16; ++k) s += red[r*16 + k];
      rowstat[r] = s;
    }
    __syncthreads();
  }

  // phase 3: hidden = P @ x_hidden via fp32 WMMA (4 col tiles x 2 K-halves)
  {
    const int wave = tid >> 5, lane = tid & 31;
    const int li = lane & 15, hi = lane >> 4;
    const int nt = wave & 3;       // hidden column tile
    const int kh = wave >> 2;      // K half
    const int col = nt*16 + li;
    const float* bcol = xhT + (size_t)col * NROWS;   // B column, contiguous in K
    v8f acc = {0.f,0.f,0.f,0.f,0.f,0.f,0.f,0.f};
    for (int kt = kh*256; kt < kh*256 + 256; ++kt) {
      const int k0 = kt*4 + 2*hi;
      v2f a   = *(const v2f*)&mw[li*2048 + k0];      // A = P fragment from LDS
      v2f bfr = *(const v2f*)&bcol[k0];              // B = xhT fragment (b64 load)
      acc = wmma4(a, bfr, acc);
    }
    if (kh == 1) {
      #pragma unroll
      for (int v = 0; v < 8; ++v) partial[nt*256 + v*32 + lane] = acc[v];
    }
    __syncthreads();
    if (kh == 0) {
      #pragma unroll
      for (int v = 0; v < 8; ++v) {
        const int row = v + 8*hi;
        const float h = (acc[v] + partial[nt*256 + v*32 + lane]) / rowstat[row]
                      + xh[(size_t)(i0 + row)*HID + col];   // softmax scale + residual
        hidbuf[row*HID + col] = h;
      }
    }
    __syncthreads();
    if (tid < 16) {
      float p = fcb[0];
      for (int c = 0; c < HID; ++c) p = fmaf(hidbuf[tid*HID + c], fcw[c], p);
      out[i0 + tid] = p;
    }
  }
}

// ---------------------------------------------------------------------------
extern "C" void kernel_launch(void* const* d_in, const int* in_sizes, int n_in,
                              void* d_out, int out_size, void* d_ws, size_t ws_size,
                              hipStream_t stream) {
  (void)in_sizes; (void)n_in; (void)out_size; (void)ws_size;
  const float* x    = (const float*)d_in[0];
  const float* rm   = (const float*)d_in[1];
  const float* Wih0 = (const float*)d_in[2];
  const float* Whh0 = (const float*)d_in[3];
  const float* bih0 = (const float*)d_in[4];
  const float* bhh0 = (const float*)d_in[5];
  const float* Wih1 = (const float*)d_in[6];
  const float* Whh1 = (const float*)d_in[7];
  const float* bih1 = (const float*)d_in[8];
  const float* bhh1 = (const float*)d_in[9];
  const float* W    = (const float*)d_in[10];
  const float* b    = (const float*)d_in[11];
  const float* fcw  = (const float*)d_in[12];
  const float* fcb  = (const float*)d_in[13];
  float* out = (float*)d_out;

  float* xh  = (float*)d_ws;                  // 2048*64 f32
  float* ai  = xh + (size_t)NROWS * HID;      // 2048 f32
  float* aj  = ai + NROWS;                    // 2048 f32
  float* xhT = aj + NROWS;                    // 64*2048 f32

  // GRU LDS: 3*12288 + 1152 + 768 weights + 4 waves * 4192 state
  const size_t gru_smem = (size_t)(3*12288 + 1152 + 768 + 4*4192) * sizeof(float);
  gru_kernel<<<NROWS/64, 128, gru_smem, stream>>>(x, Wih0, Whh0, bih0, bhh0,
                                                  Wih1, Whh1, bih1, bhh1, xh);
  attn_scalar_kernel<<<NROWS/256, 256, 0, stream>>>(xh, W, ai, aj, xhT);
  const size_t attn_smem = (size_t)(16*2048 + 256 + 16 + 16*HID + 4*256) * sizeof(float);
  attn_kernel<<<NROWS/16, 256, attn_smem, stream>>>(rm, xh, xhT, ai, aj, W, b,
                                                    fcw, fcb, out);
}